// MLPPredictor_6322191859869
// MI455X (gfx1250) — compile-verified
//
#include <hip/hip_runtime.h>

typedef float v2f __attribute__((ext_vector_type(2)));
typedef float v4f __attribute__((ext_vector_type(4)));
typedef float v8f __attribute__((ext_vector_type(8)));

#define D_IN   64
#define C_OUT  5
#define XS     132   // x/h tile row stride (floats): 132 % 64 == 4 -> conflict-free b64 A-frags
#define WFS    160   // W1f/W2f (kk,hi)-row stride (dwords): 160 % 64 == 32 -> conflict-free b64 B-frags
#define WAVES  8
#define TILE   16    // edges per wave

// Branch-free exact-GELU: gelu(x) = 0.5*(x+|x|) - 0.5*|x|*P(t)*exp(-z^2),
// z = |x|/sqrt(2), t = 1/(1+0.3275911 z)  (A&S 7.1.26, |erf err| <= 1.5e-7)
__device__ __forceinline__ float gelu_exact(float x) {
    float ax = __builtin_fabsf(x);
    float z  = ax * 0.70710678118654752f;
    float t  = __builtin_amdgcn_rcpf(__builtin_fmaf(0.3275911f, z, 1.0f));
    float p  = __builtin_fmaf(t, 1.061405429f, -1.453152027f);
    p = __builtin_fmaf(t, p, 1.421413741f);
    p = __builtin_fmaf(t, p, -0.284496736f);
    p = __builtin_fmaf(t, p, 0.254829592f);
    p *= t;
    float e  = __builtin_amdgcn_exp2f(-z * z * 1.44269504088896340f); // exp(-z^2)
    return __builtin_fmaf(-0.5f * ax, p * e, 0.5f * (x + ax));
}

__launch_bounds__(256, 1)
__global__ void mlp_edge_kernel(const float* __restrict__ ufeat,
                                const float* __restrict__ ifeat,
                                const float* __restrict__ W1,
                                const float* __restrict__ W2,
                                const float* __restrict__ Wp,
                                const int*   __restrict__ src_idx,
                                const int*   __restrict__ dst_idx,
                                float*       __restrict__ out,
                                int E)
{
    // Fragment-order weight staging: pair (W[4kk+2hi+0][n], W[4kk+2hi+1][n]) adjacent
    __shared__ float W1f[64 * WFS];          // 40 KB : rows = (kk<32, hi<2)
    __shared__ float W2f[32 * WFS];          // 20 KB : rows = (kk<16, hi<2)
    __shared__ float Wpf[32 * 32];           // 4 KB  : rows = (kk<16, hi<2), 16 padded cols
    __shared__ float Xt[WAVES * TILE * XS];  // ~66 KB per-wave 16x132 tiles

    const int tid  = threadIdx.x;
    const int lane = tid & 31;
    const int wave = tid >> 5;

    // ---- cooperative weight staging into LDS (fragment order) ----
    for (int t = tid; t < 128 * 64; t += 256) {
        int r = t >> 6, c = t & 63;          // r: K-row, c: N-col
        int kk = r >> 2, hw = (r >> 1) & 1, i = r & 1;
        W1f[(kk * 2 + hw) * WFS + c * 2 + i] = W1[t];
    }
    for (int t = tid; t < 64 * 64; t += 256) {
        int r = t >> 6, c = t & 63;
        int kk = r >> 2, hw = (r >> 1) & 1, i = r & 1;
        W2f[(kk * 2 + hw) * WFS + c * 2 + i] = W2[t];
    }
    for (int t = tid; t < 32 * 32; t += 256) {
        int row = t >> 5;                    // (kk*2+hi)
        int q = t & 31, nn = q >> 1, i = q & 1;
        int kk = row >> 1, hw = row & 1;
        int krow = kk * 4 + hw * 2 + i;      // 0..63
        Wpf[t] = (nn < C_OUT) ? Wp[krow * C_OUT + nn] : 0.0f;
    }
    __syncthreads();

    const int base = blockIdx.x * (WAVES * TILE) + wave * TILE;
    if (base >= E) return;

    float* xt = &Xt[wave * TILE * XS];
    const int m  = lane & 15;
    const int hi = lane >> 4;
    const int n  = m;

    // ---- preload edge indices: lanes 0-15 -> src, lanes 16-31 -> dst ----
    int eidx = base + m;
    if (eidx >= E) eidx = E - 1;
    const int idxv = hi ? dst_idx[eidx] : src_idx[eidx];

    // ---- gather + factor-interleave into LDS x tile [16 x 128] ----
    {
        const int c4   = lane & 15;
        const int xcol = 32 * (c4 >> 2) + 16 * hi + 4 * (c4 & 3);
        for (int e = 0; e < TILE; ++e) {
            const int su = __shfl(idxv, e, 32);
            const int sv = __shfl(idxv, 16 + e, 32);
            const float* row = hi ? (ifeat + (size_t)sv * D_IN)
                                  : (ufeat + (size_t)su * D_IN);
            v4f val = *(const v4f*)(row + 4 * c4);
            *(v4f*)(&xt[e * XS + xcol]) = val;
        }
    }

    // ---- GEMM1: h1 = x[16x128] @ W1[128x64], exact GELU ----
    v8f acc[4] = {};
    for (int kk = 0; kk < 32; ++kk) {
        v2f a = *(const v2f*)(&xt[m * XS + 4 * kk + 2 * hi]);
        const float* wrow = &W1f[(kk * 2 + hi) * WFS];
#pragma unroll
        for (int nt = 0; nt < 4; ++nt) {
            v2f b = *(const v2f*)(&wrow[(16 * nt + n) * 2]);
            acc[nt] = __builtin_amdgcn_wmma_f32_16x16x4_f32(
                false, a, false, b, (short)0, acc[nt], false, false);
        }
    }
#pragma unroll
    for (int nt = 0; nt < 4; ++nt)
#pragma unroll
        for (int i = 0; i < 8; ++i)
            acc[nt][i] = gelu_exact(acc[nt][i]);

    // h1 -> x tile (A-fragment source for GEMM2)
#pragma unroll
    for (int nt = 0; nt < 4; ++nt)
#pragma unroll
        for (int i = 0; i < 8; ++i)
            xt[(i + 8 * hi) * XS + 16 * nt + n] = acc[nt][i];

    // ---- GEMM2: h_fea = h1 @ W2[64x64] ----
    v8f acc2[4] = {};
    for (int kk = 0; kk < 16; ++kk) {
        v2f a = *(const v2f*)(&xt[m * XS + 4 * kk + 2 * hi]);
        const float* wrow = &W2f[(kk * 2 + hi) * WFS];
#pragma unroll
        for (int nt = 0; nt < 4; ++nt) {
            v2f b = *(const v2f*)(&wrow[(16 * nt + n) * 2]);
            acc2[nt] = __builtin_amdgcn_wmma_f32_16x16x4_f32(
                false, a, false, b, (short)0, acc2[nt], false, false);
        }
    }

    // h_fea -> x tile (feeds GEMM3 A-frags AND coalesced global store)
#pragma unroll
    for (int nt = 0; nt < 4; ++nt)
#pragma unroll
        for (int i = 0; i < 8; ++i)
            xt[(i + 8 * hi) * XS + 16 * nt + n] = acc2[nt][i];

    // coalesced h_fea store: 256B contiguous per edge row, b128 per lane
    float* out_h = out + (size_t)E * C_OUT;
#pragma unroll
    for (int j = 0; j < 8; ++j) {
        const int idx = lane + 32 * j;       // 0..255 = 16 rows x 16 float4-chunks
        const int er = idx >> 4, c = idx & 15;
        const int ge = base + er;
        v4f val = *(const v4f*)(&xt[er * XS + 4 * c]);
        if (ge < E)
            *(v4f*)(&out_h[(size_t)ge * 64 + 4 * c]) = val;
    }

    // ---- GEMM3: score = h_fea @ Wp (zero-padded 64x16) ----
    v8f acc3 = {};
    for (int kk = 0; kk < 16; ++kk) {
        v2f a = *(const v2f*)(&xt[m * XS + 4 * kk + 2 * hi]);
        v2f b = *(const v2f*)(&Wpf[(kk * 2 + hi) * 32 + n * 2]);
        acc3 = __builtin_amdgcn_wmma_f32_16x16x4_f32(
            false, a, false, b, (short)0, acc3, false, false);
    }
#pragma unroll
    for (int i = 0; i < 8; ++i) {
        const int ge = base + i + 8 * hi;
        if (n < C_OUT && ge < E)
            out[(size_t)ge * C_OUT + n] = acc3[i];
    }
}

extern "C" void kernel_launch(void* const* d_in, const int* in_sizes, int n_in,
                              void* d_out, int out_size, void* d_ws, size_t ws_size,
                              hipStream_t stream) {
    const float* ufeat = (const float*)d_in[0];
    const float* ifeat = (const float*)d_in[1];
    const float* W1    = (const float*)d_in[2];
    const float* W2    = (const float*)d_in[3];
    const float* Wp    = (const float*)d_in[4];
    const int*   src   = (const int*)d_in[5];
    const int*   dst   = (const int*)d_in[6];
    float*       out   = (float*)d_out;

    const int E = in_sizes[5];
    const int edges_per_block = WAVES * TILE;  // 128
    const int nblocks = (E + edges_per_block - 1) / edges_per_block;

    mlp_edge_kernel<<<nblocks, 256, 0, stream>>>(ufeat, ifeat, W1, W2, Wp,
                                                 src, dst, out, E);
}